// Ontomap_18640158065266
// MI455X (gfx1250) — compile-verified
//
#include <hip/hip_runtime.h>

typedef float v2f __attribute__((ext_vector_type(2)));
typedef float v8f __attribute__((ext_vector_type(8)));
typedef int   v4i __attribute__((ext_vector_type(4)));

#define DIM      64
#define TILE     64                  // samples staged per buffer
#define PAD      68                  // padded LDS row stride in floats
#define SEG      (TILE * PAD)        // floats per staged table
#define NBLOCKS  1024
#define THREADS  256
#define SMEM_BYTES (2 * 3 * SEG * 4) // double-buffered N/M/F staging

#define GAS __attribute__((address_space(1)))
#define LAS __attribute__((address_space(3)))

typedef GAS v4i* gv4i_p;   // global 16B-vector pointer (builtin's src type)
typedef LAS v4i* lv4i_p;   // LDS 16B-vector pointer (builtin's dst type)

#if __has_builtin(__builtin_amdgcn_global_load_async_to_lds_b128)
#define HAVE_ASYNC_LDS 1
#else
#define HAVE_ASYNC_LDS 0
#endif

__device__ __forceinline__ void async_fence() {
#if HAVE_ASYNC_LDS
#if __has_builtin(__builtin_amdgcn_s_wait_asynccnt)
    __builtin_amdgcn_s_wait_asynccnt(0);
#else
    asm volatile("s_wait_asynccnt 0x0" ::: "memory");
#endif
#endif
}

// Gather 64 rows x 64 f32 from each of the three tables into LDS.
// 4 consecutive threads own one row: each lane moves 4x16B contiguous chunks.
__device__ __forceinline__ void stage_tile(
    float* dN, float* dM, float* dF,
    const int*   __restrict__ pos_n, const int* __restrict__ pos_m,
    const int*   __restrict__ pos_f,
    const float* __restrict__ nci,   const float* __restrict__ fma,
    int base, int tid)
{
    const int row = tid >> 2;         // 0..63
    const int cb  = (tid & 3) * 16;   // starting float column (64B chunk)
    const int pn = pos_n[base + row];
    const int pm = pos_m[base + row];
    const int pf = pos_f[base + row];
    const float* sn = nci + (size_t)pn * DIM + cb;
    const float* sm = nci + (size_t)pm * DIM + cb;
    const float* sf = fma + (size_t)pf * DIM + cb;
    float* tn = dN + row * PAD + cb;
    float* tm = dM + row * PAD + cb;
    float* tf = dF + row * PAD + cb;
#if HAVE_ASYNC_LDS
    #pragma unroll
    for (int k = 0; k < 4; ++k) {
        __builtin_amdgcn_global_load_async_to_lds_b128(
            (gv4i_p)(sn + 4 * k), (lv4i_p)(tn + 4 * k), 0, 0);
        __builtin_amdgcn_global_load_async_to_lds_b128(
            (gv4i_p)(sm + 4 * k), (lv4i_p)(tm + 4 * k), 0, 0);
        __builtin_amdgcn_global_load_async_to_lds_b128(
            (gv4i_p)(sf + 4 * k), (lv4i_p)(tf + 4 * k), 0, 0);
    }
#else
    float4 vn[4], vm[4], vf[4];
    #pragma unroll
    for (int k = 0; k < 4; ++k) vn[k] = *(const float4*)(sn + 4 * k);
    #pragma unroll
    for (int k = 0; k < 4; ++k) vm[k] = *(const float4*)(sm + 4 * k);
    #pragma unroll
    for (int k = 0; k < 4; ++k) vf[k] = *(const float4*)(sf + 4 * k);
    #pragma unroll
    for (int k = 0; k < 4; ++k) {
        *(float4*)(tn + 4 * k) = vn[k];
        *(float4*)(tm + 4 * k) = vm[k];
        *(float4*)(tf + 4 * k) = vf[k];
    }
#endif
}

__global__ __launch_bounds__(THREADS) void ontomap_main(
    const int*   __restrict__ pos_n,
    const int*   __restrict__ pos_m,
    const int*   __restrict__ pos_f,
    const float* __restrict__ nci,
    const float* __restrict__ fma,
    const float* __restrict__ n2f,
    const float* __restrict__ m2f,
    float*       __restrict__ partial,
    int nSamples)
{
    extern __shared__ float smem[];        // 2 buffers x {eN, eM, eF}
    __shared__ float red[THREADS / 32];

    const int tid  = threadIdx.x;
    const int wave = tid >> 5;
    const int lane = tid & 31;
    const int lm   = lane & 15;            // sample-in-subtile / N index
    const int hi   = lane >> 4;            // K half select

    const int matSel = wave >> 2;          // 0 => n2f path, 1 => m2f path
    const int islab  = (wave & 3) * 16;    // 16 output dims owned by this wave

    // Loop-invariant A fragments: this wave's 16x64 slab of the shared matrix.
    // WMMA f32 16x16x4 A layout: lanes 0-15 -> K=0,1; lanes 16-31 -> K=2,3.
    const float* M = matSel ? m2f : n2f;
    v2f a[16];
    {
        const float* arow = M + (size_t)(islab + lm) * DIM + 2 * hi;
        #pragma unroll
        for (int kc = 0; kc < 16; ++kc)
            a[kc] = *(const v2f*)(arow + 4 * kc);
    }

    float acc_loss = 0.0f;
    const int nTiles = nSamples / TILE;

    int buf = 0;
    int t = blockIdx.x;
    if (t < nTiles)
        stage_tile(smem, smem + SEG, smem + 2 * SEG,
                   pos_n, pos_m, pos_f, nci, fma, t * TILE, tid);
    async_fence();
    __syncthreads();

    for (; t < nTiles; t += NBLOCKS) {
        // Prefetch next tile into the other buffer (overlaps with WMMA below).
        const int nxt = t + NBLOCKS;
        if (nxt < nTiles) {
            float* nb = smem + (buf ^ 1) * 3 * SEG;
            stage_tile(nb, nb + SEG, nb + 2 * SEG,
                       pos_n, pos_m, pos_f, nci, fma, nxt * TILE, tid);
        }

        const float* cur = smem + buf * 3 * SEG;
        const float* eb  = matSel ? (cur + SEG) : cur;   // eM : eN
        const float* eF  = cur + 2 * SEG;

        // 4 sub-tiles of 16 samples; D(i x sample) = A(i x K) * B(K x sample)
        #pragma unroll
        for (int st = 0; st < 4; ++st) {
            const int sample = st * 16 + lm;
            v8f acc = {};
            const float* brow = eb + sample * PAD + 2 * hi;
            #pragma unroll
            for (int kc = 0; kc < 16; ++kc) {
                v2f b = *(const v2f*)(brow + 4 * kc);    // ds_load_b64
                acc = __builtin_amdgcn_wmma_f32_16x16x4_f32(
                        false, a[kc], false, b, (short)0, acc, false, false);
            }
            // D layout: lane<16 -> M=r; lane>=16 -> M=8+r.  Matching f_e:
            // eF[sample][islab + 8*hi + r], r=0..7 contiguous -> two b128 loads.
            const float* fr = eF + sample * PAD + islab + 8 * hi;
            const float4 f0 = *(const float4*)(fr);
            const float4 f1 = *(const float4*)(fr + 4);
            const float fv[8] = {f0.x, f0.y, f0.z, f0.w, f1.x, f1.y, f1.z, f1.w};
            #pragma unroll
            for (int r = 0; r < 8; ++r) {
                const float d = acc[r] - fv[r];
                acc_loss += d * d;
            }
        }

        async_fence();      // next tile's async gathers have landed
        __syncthreads();    // and everyone is done reading the current buffer
        buf ^= 1;
    }

    // wave32 shuffle reduce, then cross-wave LDS reduce
    #pragma unroll
    for (int off = 16; off > 0; off >>= 1)
        acc_loss += __shfl_down(acc_loss, off, 32);
    if (lane == 0) red[wave] = acc_loss;
    __syncthreads();
    if (tid == 0) {
        float s = 0.0f;
        #pragma unroll
        for (int w = 0; w < THREADS / 32; ++w) s += red[w];
        partial[blockIdx.x] = s;
    }
}

// Deterministic single-block reduction of per-block partials.
__global__ __launch_bounds__(256) void ontomap_reduce(
    const float* __restrict__ partial, float* __restrict__ out)
{
    __shared__ float s[256];
    float v = 0.0f;
    for (int i = threadIdx.x; i < NBLOCKS; i += 256) v += partial[i];
    s[threadIdx.x] = v;
    __syncthreads();
    for (int off = 128; off > 0; off >>= 1) {
        if (threadIdx.x < off) s[threadIdx.x] += s[threadIdx.x + off];
        __syncthreads();
    }
    if (threadIdx.x == 0) out[0] = s[0];
}

extern "C" void kernel_launch(void* const* d_in, const int* in_sizes, int n_in,
                              void* d_out, int out_size, void* d_ws, size_t ws_size,
                              hipStream_t stream)
{
    const int*   pos_n = (const int*)d_in[0];
    const int*   pos_m = (const int*)d_in[1];
    const int*   pos_f = (const int*)d_in[2];
    const float* nci   = (const float*)d_in[3];
    const float* fma   = (const float*)d_in[4];
    const float* n2f   = (const float*)d_in[5];
    const float* m2f   = (const float*)d_in[6];
    float*       out   = (float*)d_out;
    float*       partials = (float*)d_ws;   // NBLOCKS floats (4 KB)

    const int nSamples = in_sizes[0];

    ontomap_main<<<NBLOCKS, THREADS, SMEM_BYTES, stream>>>(
        pos_n, pos_m, pos_f, nci, fma, n2f, m2f, partials, nSamples);
    ontomap_reduce<<<1, 256, 0, stream>>>(partials, out);
}